// polNormal_NoLayer_65146063946054
// MI455X (gfx1250) — compile-verified
//
#include <hip/hip_runtime.h>

// Problem constants (from the reference)
#define B_SZ   4
#define N_OUT  10240
#define SO     4
#define SI     4
#define NC     32
#define N_IN   (N_OUT * SO * SI)   // 163840
#define N_PDS  16                  // P(4) * D(4) * S(1)
#define RF     16                  // receptive field = SO*SI
#define DIST_UNIT 2.0f

typedef float v2f __attribute__((ext_vector_type(2)));
typedef float v8f __attribute__((ext_vector_type(8)));

__global__ __launch_bounds__(256) void polmix_wmma_kernel(
    const float* __restrict__ x,            // (B, N_IN, 1, 1, NC)
    const float* __restrict__ cx,           // (1, N_OUT, SO, SI)
    const float* __restrict__ cy,           // (1, N_OUT, SO, SI)
    const float* __restrict__ phis,         // (4,)
    const float* __restrict__ dists_param,  // (4,)
    const float* __restrict__ sigma_param,  // (1,)
    float* __restrict__ out)                // (B, N_OUT, 1, 4, 4, 1, NC)
{
  const int lane = threadIdx.x & 31;
  const int wave = threadIdx.x >> 5;
  const int o    = blockIdx.x * 8 + wave;     // wave-uniform
  if (o >= N_OUT) return;                     // uniform exit: EXEC stays all-1s

  const int m = lane & 15;   // mixture component index (p*4+d)
  const int h = lane >> 4;   // wave half

  // ---- mixture parameters for this lane's component m ----
  const int p = m >> 2;
  const int d = m & 3;
  const float phi = phis[p];
  const float dp  = dists_param[d];
  const float sp  = sigma_param[0];
  const float dist  = DIST_UNIT / (1.0f + __expf(-dp));      // DIST_UNIT*sigmoid
  float sigma       = DIST_UNIT / (1.0f + __expf(-sp));
  sigma             = fmaxf(sigma, 1e-10f);
  const float mux   = __cosf(phi) * dist;
  const float muy   = __sinf(phi) * dist;
  const float nhis2 = -0.5f / (sigma * sigma);

  // ---- spatial weights: lane (m,h) computes w[ij = h*8+t][m], t=0..7 ----
  // coords for this node: lane i<16 register holds c[o*16+i]; broadcast via shfl
  const float cxl = cx[o * RF + m];
  const float cyl = cy[o * RF + m];

  float q[8];
  float partial = 0.0f;
#pragma unroll
  for (int t = 0; t < 8; ++t) {
    const int ij = h * 8 + t;
    const float cxi = __shfl(cxl, ij, 32);
    const float cyi = __shfl(cyl, ij, 32);
    const float dx = cxi - mux;
    const float dy = cyi - muy;
    const float w  = __expf(nhis2 * (dx * dx + dy * dy));
    q[t] = w;
    partial += w;
  }
  // normalize over the 16 receptive positions (per component m)
  const float total = partial + __shfl_xor(partial, 16, 32);
  const float scale = 1.0f / (total + 1e-20f);
#pragma unroll
  for (int t = 0; t < 8; ++t) q[t] *= scale;

  // ---- assemble A fragments (W^T, M=pds, K=ij) per ISA 16x4 f32 A layout:
  //   V0: lane<16 -> K=4kk+0, lane>=16 -> K=4kk+2 ; V1: K=4kk+1 / 4kk+3, row M=lane%16
  // lower-half lanes own rows 0..7 in q[], upper-half own rows 8..15; swap via shfl_xor(16).
  float xq[8];
#pragma unroll
  for (int t = 0; t < 8; ++t) xq[t] = __shfl_xor(q[t], 16, 32);

  v2f A[4];
  A[0][0] = h ? xq[2] : q[0];  A[0][1] = h ? xq[3] : q[1];  // rows 0,1 | 2,3
  A[1][0] = h ? xq[6] : q[4];  A[1][1] = h ? xq[7] : q[5];  // rows 4,5 | 6,7
  A[2][0] = h ? q[2] : xq[0];  A[2][1] = h ? q[3] : xq[1];  // rows 8,9 | 10,11
  A[3][0] = h ? q[6] : xq[4];  A[3][1] = h ? q[7] : xq[5];  // rows 12,13 | 14,15

  // ---- GEMM: out(16 x 128) = A(16x16) * X(16x128), N tiled by 16 (8 chunks) ----
  const int rlo = h ? 2 : 0;       // first K-row this half supplies to B fragment
  const int col = lane & 15;       // N within chunk

#pragma unroll
  for (int nn = 0; nn < 8; ++nn) {
    const int b     = nn >> 1;
    const int c     = ((nn & 1) << 4) + col;
    const float* xb = x + ((long)b * N_IN + (long)o * RF) * NC + c;

    v8f acc = {0.f, 0.f, 0.f, 0.f, 0.f, 0.f, 0.f, 0.f};
#pragma unroll
    for (int kk = 0; kk < 4; ++kk) {
      const int r0 = 4 * kk + rlo;
      v2f Bf;
      Bf[0] = xb[(long)(r0)     * NC];   // B V0: K=4kk+0 / 4kk+2
      Bf[1] = xb[(long)(r0 + 1) * NC];   // B V1: K=4kk+1 / 4kk+3
      acc = __builtin_amdgcn_wmma_f32_16x16x4_f32(
          /*neg_a=*/false, A[kk], /*neg_b=*/false, Bf,
          /*c_mod=*/(short)0, acc, /*reuse_a=*/false, /*reuse_b=*/false);
    }

    // D layout: VGPR r -> M = r (lanes 0-15) / r+8 (lanes 16-31), N = lane%16
    float* ob = out + (((long)b * N_OUT + o) * N_PDS + (h ? 8 : 0)) * NC + c;
#pragma unroll
    for (int r = 0; r < 8; ++r) ob[(long)r * NC] = acc[r];
  }
}

extern "C" void kernel_launch(void* const* d_in, const int* in_sizes, int n_in,
                              void* d_out, int out_size, void* d_ws, size_t ws_size,
                              hipStream_t stream) {
  (void)in_sizes; (void)n_in; (void)out_size; (void)d_ws; (void)ws_size;
  const float* x   = (const float*)d_in[0];
  const float* cx  = (const float*)d_in[1];
  const float* cy  = (const float*)d_in[2];
  const float* ph  = (const float*)d_in[3];
  const float* dpp = (const float*)d_in[4];
  const float* sp  = (const float*)d_in[5];
  float* outp      = (float*)d_out;

  const int waves_per_block = 8;                       // 256 threads, wave32
  const int blocks = (N_OUT + waves_per_block - 1) / waves_per_block;  // 1280
  polmix_wmma_kernel<<<blocks, 256, 0, stream>>>(x, cx, cy, ph, dpp, sp, outp);
}